// Text2Pose_criterion_2130303779193
// MI455X (gfx1250) — compile-verified
//
#include <hip/hip_runtime.h>

// Problem constants (from the reference)
#define BB 128
#define SS 512
#define KK 137
#define DD 274          // 2*K
#define NC 136          // connections
#define TS 16           // s-rows per block
#define NTILE (SS / TS) // 32
#define NBLK (BB * NTILE) // 4096
#define PRED_ROWS (TS + 1)

// ---- gfx1250 async global->LDS copies (ASYNCcnt path) ------------------
// vdst = VGPR holding LDS byte offset, vaddr = 32-bit byte offset VGPR,
// saddr = 64-bit SGPR base (GVS addressing). Tracked by ASYNCcnt.
__device__ __forceinline__ void async_g2l_b32(unsigned lds_byte_addr,
                                              const void* gbase,
                                              unsigned gbyte_off) {
  asm volatile("global_load_async_to_lds_b32 %0, %1, %2"
               :
               : "v"(lds_byte_addr), "v"(gbyte_off), "s"(gbase)
               : "memory");
}
__device__ __forceinline__ void async_g2l_b64(unsigned lds_byte_addr,
                                              const void* gbase,
                                              unsigned gbyte_off) {
  asm volatile("global_load_async_to_lds_b64 %0, %1, %2"
               :
               : "v"(lds_byte_addr), "v"(gbyte_off), "s"(gbase)
               : "memory");
}
__device__ __forceinline__ void wait_asynccnt0() {
  asm volatile("s_wait_asynccnt 0" ::: "memory");
}

__device__ __forceinline__ unsigned lds_addr_of(const void* p) {
  // Generic pointer to LDS: low 32 bits are the workgroup-relative LDS offset.
  return (unsigned)(unsigned long long)p;
}

__global__ __launch_bounds__(256) void t2p_loss_main(
    const float* __restrict__ pred,   // [B,S,2K]   rows are 8B-aligned (1096B)
    const float* __restrict__ tgt,    // [B,2,S,K]  rows are 4B-aligned (548B)
    const int* __restrict__ len,      // [B]
    float* __restrict__ partials) {   // ws: [2*NBLK]
  __shared__ __align__(16) float lds_pred[PRED_ROWS * DD];   // rows s0-1 .. s0+TS-1 (clamped)
  __shared__ __align__(16) float lds_tgt[2 * TS * KK];       // (j*TS+rr)*KK + k
  __shared__ float red[16];

  const int blk  = blockIdx.x;
  const int b    = blk >> 5;          // NTILE = 32 tiles per batch entry
  const int tile = blk & (NTILE - 1);
  const int s0   = tile * TS;
  const int tid  = threadIdx.x;
  const int L    = len[b];

  // ---------------- async DMA stage ----------------
  // pred: 17 rows x 137 8-byte chunks (rows start 8B-aligned, length 137*8B)
  for (int i = tid; i < PRED_ROWS * (DD / 2); i += 256) {
    int rr = i / (DD / 2);
    int j  = i - rr * (DD / 2);                  // 8-byte chunk within row
    int srow = s0 - 1 + rr;
    if (srow < 0) srow = 0;                      // clamped halo (masked out later)
    unsigned goff = (unsigned)((b * SS + srow) * DD * 4 + j * 8);
    async_g2l_b64(lds_addr_of(&lds_pred[rr * DD + 2 * j]), pred, goff);
  }
  // target: 32 rows x 137 dwords (only 4B-aligned -> B32)
  for (int i = tid; i < 2 * TS * KK; i += 256) {
    int k  = i % KK;
    int t  = i / KK;                             // t = j*TS + rr
    int rr = t & (TS - 1);
    int j  = t >> 4;
    unsigned goff = (unsigned)((((b * 2 + j) * SS + (s0 + rr)) * KK + k) * 4);
    async_g2l_b32(lds_addr_of(&lds_tgt[i]), tgt, goff);
  }
  wait_asynccnt0();
  __syncthreads();

  // ---------------- compute stage ----------------
  const int wave = tid >> 5;
  const int lane = tid & 31;
  float accP = 0.0f, accB = 0.0f;

  for (int rl = wave; rl < TS; rl += 8) {        // each wave owns 2 rows
    const int r = s0 + rl;                       // global row index
    const float* prow_prev = &lds_pred[rl * DD];        // pred row r-1
    const float* prow      = &lds_pred[(rl + 1) * DD];  // pred row r
    const float* tx        = &lds_tgt[rl * KK];         // target x row r
    const float* ty        = &lds_tgt[(TS + rl) * KK];  // target y row r

    // pose L1: row r uses pred[r-1] vs tp[r]; gate 1 <= r < len (wave-uniform)
    if (r >= 1 && r < L) {
      float s = 0.0f;
      for (int d = lane; d < DD; d += 32) {
        const int k = d >> 1;
        const float tv = (d & 1) ? ty[k] : tx[k];
        s += fabsf(prow_prev[d] - tv);
      }
      accP += s * (1.0f / (float)DD);
    }
    // bone MSE: gate r < len (wave-uniform)
    if (r < L) {
      float s = 0.0f;
      for (int c = lane; c < NC; c += 32) {
        const float ex = (prow[2 * c + 2] - prow[2 * c])     - (tx[c + 1] - tx[c]);
        const float ey = (prow[2 * c + 3] - prow[2 * c + 1]) - (ty[c + 1] - ty[c]);
        s += 0.5f * (ex * ex + ey * ey);
      }
      accB += s * (1.0f / (136.0f + 1e-8f));
    }
  }

  // ---------------- deterministic block reduction ----------------
  for (int off = 16; off; off >>= 1) {
    accP += __shfl_xor(accP, off, 32);
    accB += __shfl_xor(accB, off, 32);
  }
  if (lane == 0) { red[wave] = accP; red[8 + wave] = accB; }
  __syncthreads();
  if (tid == 0) {
    float p = 0.0f, q = 0.0f;
    for (int w = 0; w < 8; ++w) { p += red[w]; q += red[8 + w]; }
    partials[blk]        = p;
    partials[NBLK + blk] = q;
  }
}

__global__ __launch_bounds__(256) void t2p_loss_final(
    const float* __restrict__ partials, const int* __restrict__ len,
    float* __restrict__ out) {
  __shared__ float sp[256], sb[256];
  __shared__ int   sl[256];
  const int tid = threadIdx.x;

  float p = 0.0f, q = 0.0f;
  for (int i = tid; i < NBLK; i += 256) {
    p += partials[i];
    q += partials[NBLK + i];
  }
  int l = (tid < BB) ? len[tid] : 0;
  sp[tid] = p; sb[tid] = q; sl[tid] = l;
  __syncthreads();
  for (int off = 128; off; off >>= 1) {
    if (tid < off) {
      sp[tid] += sp[tid + off];
      sb[tid] += sb[tid + off];
      sl[tid] += sl[tid + off];
    }
    __syncthreads();
  }
  if (tid == 0) {
    const float sum_len  = (float)sl[0];          // sum(mask)
    const float pose_den = sum_len - (float)BB;   // sum(len-1) = sum(pose_mask)
    const float pose = sp[0] / pose_den;
    const float bone = sb[0] / sum_len;
    out[0] = 1.0f * pose + 0.1f * bone;           // total (POSE_W=1.0, BONE_W=0.1)
    out[1] = pose;
    out[2] = bone;
  }
}

extern "C" void kernel_launch(void* const* d_in, const int* in_sizes, int n_in,
                              void* d_out, int out_size, void* d_ws, size_t ws_size,
                              hipStream_t stream) {
  (void)in_sizes; (void)n_in; (void)out_size; (void)ws_size;
  const float* pred = (const float*)d_in[0];
  const float* tgt  = (const float*)d_in[1];
  const int*   len  = (const int*)d_in[2];
  float* ws  = (float*)d_ws;   // needs 2*NBLK*4 = 32 KB
  float* out = (float*)d_out;

  t2p_loss_main<<<NBLK, 256, 0, stream>>>(pred, tgt, len, ws);
  t2p_loss_final<<<1, 256, 0, stream>>>(ws, len, out);
}